// Model_11888469476170
// MI455X (gfx1250) — compile-verified
//
#include <hip/hip_runtime.h>

// Problem dims (from reference): X (4,64,32,128,128) f32, W (64,256) f32, out (4,256) f32
#define N_   4
#define C_   64
#define D_   32
#define H_   128
#define Wd_  128
#define M_   256
#define SLICES (N_ * C_)               // 256 (n,c) slices
#define DHW    (D_ * H_ * Wd_)         // 524288 elements / slice (2 MiB)
#define V4_PER_SLICE (DHW / 4)         // 131072 float4 / slice
#define BPS   32                       // blocks per slice
#define TPB   256                      // 8 waves (wave32)
#define V4_PER_BLOCK (V4_PER_SLICE / BPS)   // 4096
#define ITERS (V4_PER_BLOCK / TPB)          // 16 float4 per thread

typedef float v2f __attribute__((ext_vector_type(2)));
typedef float v4f __attribute__((ext_vector_type(4)));
typedef float v8f __attribute__((ext_vector_type(8)));

// ---------------------------------------------------------------------------
// Kernel 1: streaming triple-reduction over X. Memory-bound: 512 MiB read once.
// Non-temporal b128 loads (data is single-use; don't thrash 192MB L2).
// Deterministic: fixed partition, fixed-order LDS tree, per-block partials.
// ---------------------------------------------------------------------------
__global__ void __launch_bounds__(TPB)
reduce3_kernel(const float* __restrict__ X, float* __restrict__ partials) {
    const int slice = blockIdx.y;     // (n*64 + c)
    const int chunk = blockIdx.x;     // 0..BPS-1
    const int tid   = threadIdx.x;

    const v4f* __restrict__ Xs = (const v4f*)(X + (size_t)slice * DHW);

    float s = 0.f, s2 = 0.f, sw = 0.f;
#pragma unroll
    for (int i = 0; i < ITERS; ++i) {
        const int idx4 = chunk * V4_PER_BLOCK + i * TPB + tid; // coalesced
        const v4f v = __builtin_nontemporal_load(&Xs[idx4]);

        // element index -> (h, w0). elem = idx4*4; rows are 128 (pow2, mult of 4)
        const int h  = (idx4 >> 5) & 127;        // (elem >> 7) & 127
        const int w0 = (idx4 & 31) << 2;         // elem & 127 (multiple of 4)

        const float wh = 1.0f + (h == 1 ? 1.0f : 0.0f) + (h == 126 ? 1.0f : 0.0f);

        const float vs = (v.x + v.y) + (v.z + v.w);
        s += vs;
        s2 = fmaf(v.x, v.x, s2);
        s2 = fmaf(v.y, v.y, s2);
        s2 = fmaf(v.z, v.z, s2);
        s2 = fmaf(v.w, v.w, s2);

        // reflection weights in w: w==1 only when w0==0 (comp 1);
        // w==126 only when w0==124 (comp 2). All others weight 1.
        float extra = 0.f;
        if (w0 == 0)   extra += v.y;
        if (w0 == 124) extra += v.z;
        sw = fmaf(wh, vs + extra, sw);
    }

    __shared__ float rs[TPB], rs2[TPB], rsw[TPB];
    rs[tid] = s; rs2[tid] = s2; rsw[tid] = sw;
    __syncthreads();
#pragma unroll
    for (int off = TPB / 2; off > 0; off >>= 1) {
        if (tid < off) {
            rs[tid]  += rs[tid + off];
            rs2[tid] += rs2[tid + off];
            rsw[tid] += rsw[tid + off];
        }
        __syncthreads();
    }
    if (tid == 0) {
        const int p = slice * BPS + chunk;
        partials[p * 3 + 0] = rs[0];
        partials[p * 3 + 1] = rs2[0];
        partials[p * 3 + 2] = rsw[0];
    }
}

// ---------------------------------------------------------------------------
// Kernel 2: combine BPS partials per slice -> descriptor value (256 floats).
// Xdesc = rsqrt(var+eps) * ( Sw/(D*130*130) - mean )
// ---------------------------------------------------------------------------
__global__ void __launch_bounds__(BPS)
finalize_desc_kernel(const float* __restrict__ partials, float* __restrict__ desc) {
    const int slice = blockIdx.x;
    const int tid   = threadIdx.x;   // 0..31, one partial each

    __shared__ float rs[BPS], rs2[BPS], rsw[BPS];
    const int p = slice * BPS + tid;
    rs[tid]  = partials[p * 3 + 0];
    rs2[tid] = partials[p * 3 + 1];
    rsw[tid] = partials[p * 3 + 2];
    __syncthreads();
#pragma unroll
    for (int off = BPS / 2; off > 0; off >>= 1) {
        if (tid < off) {
            rs[tid]  += rs[tid + off];
            rs2[tid] += rs2[tid + off];
            rsw[tid] += rsw[tid + off];
        }
        __syncthreads();
    }
    if (tid == 0) {
        const float inv  = 1.0f / (float)DHW;
        const float mean = rs[0] * inv;
        const float var  = fmaf(-mean, mean, rs2[0] * inv);
        const float rsq  = rsqrtf(var + 1e-5f);
        const float wmean = rsw[0] / (float)(D_ * 130 * 130);
        desc[slice] = rsq * (wmean - mean);
    }
}

// ---------------------------------------------------------------------------
// Kernel 3: out(4x256) = desc(4x64) @ W(64x256) via V_WMMA_F32_16X16X4_F32.
// 16 waves, one 16-wide N tile each; M tile 16 with rows 4..15 zero-padded;
// K=64 -> 16 WMMA steps of K=4. Fragment layouts per ISA 7.12.2:
//   A 16x4 f32: lanes0-15=M, v0:{K=0|K=2}, v1:{K=1|K=3}  (hi half lanes -> K+2)
//   B 4x16 f32: row (K) striped across lanes within a VGPR, symmetric to A
//   C/D 16x16:  VGPR r = rows r / r+8; lanes 0-15 / 16-31
// ---------------------------------------------------------------------------
__global__ void __launch_bounds__(32)
proj_wmma_kernel(const float* __restrict__ desc, const float* __restrict__ Wm,
                 float* __restrict__ out) {
    const int ntile = blockIdx.x;          // 0..15 -> columns [16*ntile, +16)
    const int lane  = threadIdx.x;         // 0..31
    const int mn    = lane & 15;           // M for A, N for B
    const int kb    = (lane >> 4) << 1;    // 0 (lanes 0-15) or 2 (lanes 16-31)

    v8f c = {};
#pragma unroll
    for (int k0 = 0; k0 < C_; k0 += 4) {
        v2f a, b;
        const int mrow = (mn < N_) ? mn : 0;          // avoid OOB; zero below
        const float am0 = desc[mrow * C_ + k0 + kb + 0];
        const float am1 = desc[mrow * C_ + k0 + kb + 1];
        a.x = (mn < N_) ? am0 : 0.0f;
        a.y = (mn < N_) ? am1 : 0.0f;
        b.x = Wm[(k0 + kb + 0) * M_ + ntile * 16 + mn];
        b.y = Wm[(k0 + kb + 1) * M_ + ntile * 16 + mn];
        // 8 args: (neg_a, A, neg_b, B, c_mod, C, reuse_a, reuse_b)
        c = __builtin_amdgcn_wmma_f32_16x16x4_f32(false, a, false, b,
                                                  (short)0, c, false, false);
    }
    if (lane < 16) {
#pragma unroll
        for (int r = 0; r < N_; ++r)       // only rows 0..3 are valid output
            out[r * M_ + ntile * 16 + lane] = c[r];
    }
}

// ---------------------------------------------------------------------------
extern "C" void kernel_launch(void* const* d_in, const int* in_sizes, int n_in,
                              void* d_out, int out_size, void* d_ws, size_t ws_size,
                              hipStream_t stream) {
    const float* X  = (const float*)d_in[0];   // (4,64,32,128,128) f32
    const float* Wm = (const float*)d_in[1];   // (64,256) f32
    float* out      = (float*)d_out;           // (4,256) f32

    float* partials = (float*)d_ws;                    // SLICES*BPS*3 floats (96 KB)
    float* desc     = partials + SLICES * BPS * 3;     // 256 floats

    dim3 g1(BPS, SLICES);
    reduce3_kernel<<<g1, TPB, 0, stream>>>(X, partials);
    finalize_desc_kernel<<<SLICES, BPS, 0, stream>>>(partials, desc);
    proj_wmma_kernel<<<16, 32, 0, stream>>>(desc, Wm, out);
}